// Interaction_MM4d_layer_36086315221300
// MI455X (gfx1250) — compile-verified
//
#include <hip/hip_runtime.h>
#include <math.h>

// ---------------------------------------------------------------------------
// MI455X / gfx1250 implementation of Interaction_MM4d_layer.
//  - one wave32 per Mamba sequence; all matmuls via V_WMMA_F32_16X16X4_F32
//  - SSM scan keeps h[64] + A_row[64] in VGPRs (lane = channel d)
//  - deterministic segment-mean aggregation (no float atomics)
// ---------------------------------------------------------------------------

typedef __attribute__((ext_vector_type(2))) float v2f;
typedef __attribute__((ext_vector_type(8))) float v8f;

struct Mm4dParams {
  const float *A_log, *D, *b_in, *b_out, *conv_b, *conv_w, *dt_b, *dt_w,
              *in_w, *ln_b, *ln_g, *out_w, *w_in, *w_out, *xproj_w;
};

// LDS workspace for one sequence (padded strides: 66 / 34 -> no bank conflicts)
struct SeqShared {
  float bufA[32 * 66];  // Xin (64 wide)  -> later Bm (32x64)
  float bufC[32 * 66];  // Cm (32x64)
  float x   [32 * 34];  // x = Xin@w_in   -> later delta (32x32)
  float xc  [32 * 34];  // conv+silu out  -> later final LN output
  float z   [32 * 34];  // gate
  float ys  [32 * 34];  // scan out -> yz
  float m   [32 * 34];  // mamba out -> silu(mamba)
  float dt  [32 * 2];   // dt projection (r = 2)
};

__device__ __forceinline__ float silu_f(float x) {
  return x / (1.0f + __expf(-x));
}
__device__ __forceinline__ float softplus_f(float x) {
  return fmaxf(x, 0.0f) + log1pf(__expf(-fabsf(x)));
}

// C(32 x 16*ntiles) = A(32 x K) @ B(K x 16*ntiles) [+ bias], one wave32.
// A: LDS row-major (lda), B: global row-major (ldb), C: LDS row-major (ldc).
__device__ __forceinline__ void wave_gemm(const float* __restrict__ A, int lda,
                                          const float* __restrict__ B, int ldb,
                                          float* __restrict__ C, int ldc,
                                          int ntiles, int K,
                                          const float* __restrict__ bias) {
  const int lane = threadIdx.x;
  const int half = lane >> 4;   // 0 | 1
  const int l16  = lane & 15;
  for (int mi = 0; mi < 2; ++mi) {
    const int row = mi * 16 + l16;
    for (int ni = 0; ni < ntiles; ++ni) {
      const int col = ni * 16 + l16;
      v8f acc = {0.f, 0.f, 0.f, 0.f, 0.f, 0.f, 0.f, 0.f};
      for (int k0 = 0; k0 < K; k0 += 4) {
        const int ka = k0 + 2 * half;
        v2f av, bv;
        av.x = A[row * lda + ka];          // A 16x4 frag: VGPR0=K(0|2), VGPR1=K(1|3)
        av.y = A[row * lda + ka + 1];
        bv.x = B[ka * ldb + col];          // B 4x16 frag mirrored
        bv.y = B[(ka + 1) * ldb + col];
        acc = __builtin_amdgcn_wmma_f32_16x16x4_f32(
            false, av, false, bv, (short)0, acc, false, false);
      }
      const float bval = bias ? bias[col] : 0.0f;
#pragma unroll
      for (int v = 0; v < 8; ++v) {        // C/D: M = v + 8*half
        const int r = mi * 16 + v + 8 * half;
        C[r * ldc + col] = acc[v] + bval;
      }
    }
  }
}

// Full _mm4d pipeline for one sequence. Caller fills sh.bufA with Xin (32x64,
// stride 66). Result (post-LayerNorm, 32x32) left in sh.xc (stride 34).
__device__ void mm4d_seq(SeqShared& sh, const Mm4dParams& P) {
  const int lane = threadIdx.x;

  // A) x = Xin @ w_in + b_in                          (32x64)@(64x32)
  wave_gemm(sh.bufA, 66, P.w_in, 32, sh.x, 34, 2, 64, P.b_in);
  __syncthreads();
  // B) xz = x @ in_w  -> bufA (32x64)                 (32x32)@(32x64)
  wave_gemm(sh.x, 34, P.in_w, 64, sh.bufA, 66, 4, 32, nullptr);
  __syncthreads();
  // C) causal depthwise conv (D_CONV=2) + SiLU; split off gate z
  {
    const float cw0 = P.conv_w[lane * 2 + 0];
    const float cw1 = P.conv_w[lane * 2 + 1];
    const float cb  = P.conv_b[lane];
    float prev = 0.0f;
    for (int t = 0; t < 32; ++t) {
      const float cur = sh.bufA[t * 66 + lane];
      sh.z[t * 34 + lane]  = sh.bufA[t * 66 + 32 + lane];
      sh.xc[t * 34 + lane] = silu_f(prev * cw0 + cur * cw1 + cb);
      prev = cur;
    }
  }
  __syncthreads();
  // D) dbc = xc @ xproj_w : Bm -> bufA, Cm -> bufC, dt -> sh.dt
  wave_gemm(sh.xc, 34, P.xproj_w + 2,  130, sh.bufA, 66, 4, 32, nullptr);
  wave_gemm(sh.xc, 34, P.xproj_w + 66, 130, sh.bufC, 66, 4, 32, nullptr);
  {
    float d0 = 0.f, d1 = 0.f;                 // lane == t
    for (int k = 0; k < 32; ++k) {
      const float xv = sh.xc[lane * 34 + k];
      d0 += xv * P.xproj_w[k * 130 + 0];
      d1 += xv * P.xproj_w[k * 130 + 1];
    }
    sh.dt[lane * 2 + 0] = d0;
    sh.dt[lane * 2 + 1] = d1;
  }
  __syncthreads();
  // E) delta = softplus(dt @ dt_w + dt_b)  -> sh.x
  {
    const float w0 = P.dt_w[0 * 32 + lane];
    const float w1 = P.dt_w[1 * 32 + lane];
    const float bd = P.dt_b[lane];
    for (int t = 0; t < 32; ++t) {
      const float v = sh.dt[t * 2 + 0] * w0 + sh.dt[t * 2 + 1] * w1 + bd;
      sh.x[t * 34 + lane] = softplus_f(v);
    }
  }
  __syncthreads();
  // F) selective scan: lane = d, h[s]/A[s] in VGPRs
  {
    float Arow[64], h[64];
#pragma unroll
    for (int s = 0; s < 64; ++s) {
      Arow[s] = -__expf(P.A_log[lane * 64 + s]);
      h[s] = 0.0f;
    }
    for (int t = 0; t < 32; ++t) {
      const float del = sh.x[t * 34 + lane];
      const float dtx = del * sh.xc[t * 34 + lane];
      float acc = 0.0f;
#pragma unroll
      for (int s = 0; s < 64; ++s) {
        const float bmv = sh.bufA[t * 66 + s];   // LDS broadcast
        const float cmv = sh.bufC[t * 66 + s];
        h[s] = h[s] * __expf(del * Arow[s]) + dtx * bmv;
        acc += h[s] * cmv;
      }
      sh.ys[t * 34 + lane] = acc;
    }
  }
  __syncthreads();
  // G) y = (ys + xc*D) * silu(z); m = y @ out_w
  {
    const float Dd = P.D[lane];
    for (int t = 0; t < 32; ++t) {
      const float yv = sh.ys[t * 34 + lane] + sh.xc[t * 34 + lane] * Dd;
      sh.ys[t * 34 + lane] = yv * silu_f(sh.z[t * 34 + lane]);
    }
  }
  __syncthreads();
  wave_gemm(sh.ys, 34, P.out_w, 32, sh.m, 34, 2, 32, nullptr);
  __syncthreads();
  // H) outer silu, then @ w_out + b_out -> sh.xc
  for (int t = 0; t < 32; ++t)
    sh.m[t * 34 + lane] = silu_f(sh.m[t * 34 + lane]);
  __syncthreads();
  wave_gemm(sh.m, 34, P.w_out, 32, sh.xc, 34, 2, 32, P.b_out);
  __syncthreads();
  // LayerNorm over all 1024 elements, then ln_g / ln_b
  {
    float s = 0.f, s2 = 0.f;
    for (int t = 0; t < 32; ++t) {
      const float v = sh.xc[t * 34 + lane];
      s += v; s2 += v * v;
    }
#pragma unroll
    for (int off = 16; off > 0; off >>= 1) {
      s  += __shfl_xor(s,  off, 32);
      s2 += __shfl_xor(s2, off, 32);
    }
    const float mu  = s * (1.0f / 1024.0f);
    const float var = s2 * (1.0f / 1024.0f) - mu * mu;
    const float inv = 1.0f / sqrtf(var + 1e-5f);
    for (int t = 0; t < 32; ++t) {
      const float v = (sh.xc[t * 34 + lane] - mu) * inv;
      sh.xc[t * 34 + lane] = v * P.ln_g[t * 32 + lane] + P.ln_b[t * 32 + lane];
    }
  }
  __syncthreads();
}

// ---- edge layer: seq = e*2 + b ; Xin = [x[row], x[col]] -------------------
__global__ void __launch_bounds__(32)
edge_mm4d_kernel(const float* __restrict__ nf, const int* __restrict__ ei,
                 const float* __restrict__ emask, Mm4dParams P,
                 float* __restrict__ msg_e, int E) {
  __shared__ SeqShared sh;
  const int lane = threadIdx.x;           // lane == t for load, o for store
  const int seq = blockIdx.x;
  const int e = seq >> 1, bb = seq & 1;
  const int cn = ei[e];                   // col = edge_index[0]
  const int rn = ei[E + e];               // row = edge_index[1]
  for (int d = 0; d < 64; ++d) {
    const int node = (d < 32) ? rn : cn;
    const int dd   = (d < 32) ? d : d - 32;
    sh.bufA[lane * 66 + d] = nf[bb * 32768 + dd * 1024 + node * 32 + lane];
  }
  __syncthreads();
  mm4d_seq(sh, P);
  const float mk = emask[bb * E + e];
  float* dst = msg_e + (size_t)e * 2048 + (size_t)bb * 1024;
  for (int t = 0; t < 32; ++t)
    dst[t * 32 + lane] = sh.xc[t * 34 + lane] * mk;   // msg_e[e,b,t,o]
}

// ---- deterministic segment mean over col ----------------------------------
__global__ void __launch_bounds__(256)
agg_kernel(const float* __restrict__ msg_e, const int* __restrict__ ei,
           float* __restrict__ x_adj, int E) {
  const int n = blockIdx.x;
  const int tid = threadIdx.x;            // 256 threads cover 2048 elems (x8)
  float acc[8] = {0.f, 0.f, 0.f, 0.f, 0.f, 0.f, 0.f, 0.f};
  int cnt = 0;
  for (int e = 0; e < E; ++e) {
    if (ei[e] == n) {                     // uniform branch (scalar load)
      ++cnt;
      const float* src = msg_e + (size_t)e * 2048;
#pragma unroll
      for (int j = 0; j < 8; ++j) acc[j] += src[tid + j * 256];
    }
  }
  const float c = fmaxf((float)cnt, 1.0f);
#pragma unroll
  for (int j = 0; j < 8; ++j)
    x_adj[(size_t)n * 2048 + tid + j * 256] = acc[j] / c;
}

// ---- node layer: seq = n*2 + b ; Xin = [x, x_adj]; residual + mask --------
__global__ void __launch_bounds__(32)
node_mm4d_kernel(const float* __restrict__ nf, const float* __restrict__ xadj,
                 const float* __restrict__ nmask, Mm4dParams P,
                 float* __restrict__ out) {
  __shared__ SeqShared sh;
  const int lane = threadIdx.x;           // lane == t
  const int seq = blockIdx.x;
  const int n = seq >> 1, bb = seq & 1;
  for (int d = 0; d < 32; ++d)
    sh.bufA[lane * 66 + d] = nf[bb * 32768 + d * 1024 + n * 32 + lane];
  for (int d = 0; d < 32; ++d)
    sh.bufA[lane * 66 + 32 + d] = xadj[(size_t)n * 2048 + bb * 1024 + lane * 32 + d];
  __syncthreads();
  mm4d_seq(sh, P);
  const float mk = nmask[bb * 32 + n];
  for (int o = 0; o < 32; ++o) {
    const size_t idx = (size_t)bb * 32768 + (size_t)o * 1024 + n * 32 + lane;
    out[idx] = (nf[idx] + sh.xc[lane * 34 + o]) * mk;   // out[b,o,n,t]
  }
}

// ---------------------------------------------------------------------------
static Mm4dParams params_from_leaves(void* const* p) {
  Mm4dParams P;   // jax pytree dict order = alphabetical
  P.A_log  = (const float*)p[0];  P.D      = (const float*)p[1];
  P.b_in   = (const float*)p[2];  P.b_out  = (const float*)p[3];
  P.conv_b = (const float*)p[4];  P.conv_w = (const float*)p[5];
  P.dt_b   = (const float*)p[6];  P.dt_w   = (const float*)p[7];
  P.in_w   = (const float*)p[8];  P.ln_b   = (const float*)p[9];
  P.ln_g   = (const float*)p[10]; P.out_w  = (const float*)p[11];
  P.w_in   = (const float*)p[12]; P.w_out  = (const float*)p[13];
  P.xproj_w= (const float*)p[14];
  return P;
}
static Mm4dParams params_from_blob(const float* b) {
  Mm4dParams P;   // alphabetical packing, n_in = 64
  P.A_log = b;          P.D     = b + 2048;  P.b_in  = b + 2080;
  P.b_out = b + 2112;   P.conv_b= b + 2144;  P.conv_w= b + 2176;
  P.dt_b  = b + 2240;   P.dt_w  = b + 2272;  P.in_w  = b + 2336;
  P.ln_b  = b + 4384;   P.ln_g  = b + 5408;  P.out_w = b + 6432;
  P.w_in  = b + 7456;   P.w_out = b + 9504;  P.xproj_w = b + 10528;
  return P;
}

extern "C" void kernel_launch(void* const* d_in, const int* in_sizes, int n_in,
                              void* d_out, int out_size, void* d_ws, size_t ws_size,
                              hipStream_t stream) {
  (void)out_size; (void)ws_size;
  const float* nf    = (const float*)d_in[0];   // node_feature (2,32,32,32)
  const float* nmask = (const float*)d_in[2];   // nodes_mask   (2,32)
  const float* emask = (const float*)d_in[3];   // edges_mask   (2,E)
  const int*   ei    = (const int*)d_in[4];     // edge_index   (2,E)
  const int E = in_sizes[4] / 2;

  Mm4dParams EP, NP;
  if (n_in >= 35) {
    EP = params_from_leaves(d_in + 5);
    NP = params_from_leaves(d_in + 20);
  } else {
    EP = params_from_blob((const float*)d_in[5]);
    NP = params_from_blob((const float*)d_in[6]);
  }

  float* out   = (float*)d_out;        // 65536 floats
  float* msg_e = out + 65536;          // E*2048 floats (output #2, reused as temp)
  float* x_adj = (float*)d_ws;         // 32*2048 floats scratch

  edge_mm4d_kernel<<<2 * E, 32, 0, stream>>>(nf, ei, emask, EP, msg_e, E);
  agg_kernel      <<<32, 256, 0, stream>>>(msg_e, ei, x_adj, E);
  node_mm4d_kernel<<<64, 32, 0, stream>>>(nf, x_adj, nmask, NP, out);
}